// LengthRegulator_12189117186070
// MI455X (gfx1250) — compile-verified
//
#include <hip/hip_runtime.h>
#include <math.h>
#include <stdint.h>

// ---------------------------------------------------------------------------
// LengthRegulator / duration-predictor for MI455X (gfx1250, wave32).
//   conv1d(K=3) x2 as WMMA f32 GEMMs with async-LDS-staged A (halo) and
//   repacked coalesced B; fused LN+ReLU (+linear head); single-block LDS
//   cumsum scan; binary-search expand/gather.
// ---------------------------------------------------------------------------

#define S_LEN 4096
#define D_DIM 512          // input channels  (== INNER)
#define N_DIM 512          // output channels (== INNER)
#define KW    3
#define MAX_OUT (4 * S_LEN)

#define M_BLK 32           // output rows per block (2 M-tiles per wave)
#define A_ROWS (M_BLK + 2) // halo rows staged in LDS
#define A_STRIDE 516       // LDS row stride in floats (bank-conflict spread)
#define A_CHUNKS (A_ROWS * D_DIM / 1024)   // 17 float4-chunks per thread

#define AS_GLOBAL __attribute__((address_space(1)))
#define AS_LOCAL  __attribute__((address_space(3)))

typedef int v4i_vec __attribute__((vector_size(16)));   // matches builtin's pointee

#if __has_builtin(__builtin_amdgcn_global_load_async_to_lds_b128)
#define USE_ASYNC_LDS 1
#else
#define USE_ASYNC_LDS 0
#endif

#if __has_builtin(__builtin_amdgcn_s_wait_asynccnt)
#define WAIT_ASYNC() __builtin_amdgcn_s_wait_asynccnt(0)
#else
#define WAIT_ASYNC() asm volatile("s_wait_asynccnt 0x0" ::: "memory")
#endif

typedef __attribute__((ext_vector_type(2))) float v2f;
typedef __attribute__((ext_vector_type(8))) float v8f;

// ---------------------------------------------------------------------------
// Repack conv weights W[n][kk][ks] -> Wt[ks][kk][n] so that B-matrix fragment
// loads are coalesced along n (lanes = consecutive output channels).
// ---------------------------------------------------------------------------
__global__ __launch_bounds__(256)
void repack_w_kernel(const float* __restrict__ W, float* __restrict__ Wt)
{
    const int idx = blockIdx.x * 256 + threadIdx.x;   // over 3*512*512
    const int n   = idx & (N_DIM - 1);
    const int kk  = (idx >> 9) & (D_DIM - 1);
    const int ks  = idx >> 18;
    Wt[idx] = W[(size_t)n * (D_DIM * KW) + (size_t)kk * KW + ks];
}

// ---------------------------------------------------------------------------
// Conv1d (pad=1, K=3) as GEMM: Y[s,o] = bias[o] + sum_k sum_i X[s+k-1,i]*W[o,i,k]
//
// Block: 256 threads (8 waves); tile = 32 M rows x 256 output channels.
// A rows m0-1..m0+32 staged in LDS via GLOBAL_LOAD_ASYNC_TO_LDS_B128 (halo
// covers all 3 shifts; edge rows zero-fixed after s_wait_asynccnt so the
// inner loop is branchless). Each wave owns a 2x2 grid of 16x16 accumulators:
// 2 ds_load_b64 A-fragments + 4 coalesced-B fragments feed 4
// V_WMMA_F32_16X16X4_F32 per K-step (1 B-load per WMMA).
//
// Fragment layout (ISA 7.12.2, 32-bit A 16x4):
//   lanes 0-15 : v[0]=A[M=lane][K=0], v[1]=A[M=lane][K=1]
//   lanes 16-31: v[0]=A[M=lane-16][K=2], v[1]=A[M=lane-16][K=3]
// B (4x16) symmetric; C/D 16x16: VGPR g -> M = g + 8*(lane>=16), N = lane&15.
// ---------------------------------------------------------------------------
__global__ __launch_bounds__(256)
void conv1d_gemm_wmma(const float* __restrict__ X,    // [S_LEN, D_DIM]
                      const float* __restrict__ Wt,   // [KW, D_DIM, N_DIM]
                      const float* __restrict__ bias, // [N_DIM]
                      float* __restrict__ Y)          // [S_LEN, N_DIM]
{
    __shared__ float la[A_ROWS * A_STRIDE];           // ~68.6 KB

    const int tid  = threadIdx.x;
    const int lane = tid & 31;
    const int wv   = tid >> 5;
    const int m0   = (blockIdx.x >> 1) * M_BLK;       // 128 M-blocks
    const int nh   = (blockIdx.x & 1) * 256;          // N half: [0,256) or [256,512)
    const int half = lane >> 4;                       // 0: lanes 0-15, 1: 16-31
    const int lm   = lane & 15;

    // ---- Stage A halo rows m0-1 .. m0+M_BLK into LDS ----
#if USE_ASYNC_LDS
#pragma unroll
    for (int chunk = 0; chunk < A_CHUNKS; ++chunk) {
        const int e = chunk * 1024 + tid * 4;
        const int r = e >> 9;
        const int c = e & 511;
        int row = m0 - 1 + r;                          // clamp; edges fixed below
        row = row < 0 ? 0 : (row >= S_LEN ? S_LEN - 1 : row);
        __builtin_amdgcn_global_load_async_to_lds_b128(
            (AS_GLOBAL v4i_vec*)(uintptr_t)(X + (size_t)row * D_DIM + c),
            (AS_LOCAL  v4i_vec*)(uintptr_t)(&la[r * A_STRIDE + c]), 0, 0);
    }
    WAIT_ASYNC();
    // zero-fix OOB halo rows (only the first/last M-block ever hit these)
    if (m0 == 0 && tid < 128)
        *(float4*)(&la[0 * A_STRIDE + tid * 4]) = make_float4(0.f, 0.f, 0.f, 0.f);
    if (m0 + M_BLK >= S_LEN && tid < 128)
        *(float4*)(&la[(A_ROWS - 1) * A_STRIDE + tid * 4]) = make_float4(0.f, 0.f, 0.f, 0.f);
#else
#pragma unroll
    for (int chunk = 0; chunk < A_CHUNKS; ++chunk) {
        const int e   = chunk * 1024 + tid * 4;
        const int r   = e >> 9;
        const int c   = e & 511;
        const int row = m0 - 1 + r;
        float4 v = make_float4(0.f, 0.f, 0.f, 0.f);
        if (row >= 0 && row < S_LEN)
            v = *(const float4*)(X + (size_t)row * D_DIM + c);
        *(float4*)(&la[r * A_STRIDE + c]) = v;
    }
#endif
    __syncthreads();

    // ---- Accumulate: wave covers channels [nh + wv*32, nh + wv*32 + 32) ----
    const int nbase = nh + wv * 32 + lm;              // lane's column in N-tile 0
    v8f acc00 = {0.f,0.f,0.f,0.f,0.f,0.f,0.f,0.f};   // [M-tile][N-tile]
    v8f acc01 = acc00, acc10 = acc00, acc11 = acc00;

    for (int ks = 0; ks < KW; ++ks) {
        const float* ar0 = &la[(lm + ks) * A_STRIDE];        // M-tile 0 input row
        const float* ar1 = &la[(lm + 16 + ks) * A_STRIDE];   // M-tile 1 input row
        const float* wp  = Wt + (size_t)ks * (D_DIM * N_DIM) + nbase;
        __builtin_prefetch(wp, 0, 3);

#pragma unroll 2
        for (int kk = 0; kk < D_DIM; kk += 4) {
            const int ka = kk + 2 * half;             // this lane's K pair
            const float2 a0v = *(const float2*)(ar0 + ka);   // ds_load_b64
            const float2 a1v = *(const float2*)(ar1 + ka);   // ds_load_b64
            v2f a0; a0.x = a0v.x; a0.y = a0v.y;
            v2f a1; a1.x = a1v.x; a1.y = a1v.y;

            const float* wr0 = wp + (size_t)ka * N_DIM;      // row ka
            const float* wr1 = wr0 + N_DIM;                  // row ka+1
            v2f b0; b0.x = wr0[ 0]; b0.y = wr1[ 0];
            v2f b1; b1.x = wr0[16]; b1.y = wr1[16];

            acc00 = __builtin_amdgcn_wmma_f32_16x16x4_f32(false, a0, false, b0, (short)0, acc00, false, false);
            acc01 = __builtin_amdgcn_wmma_f32_16x16x4_f32(false, a0, false, b1, (short)0, acc01, false, false);
            acc10 = __builtin_amdgcn_wmma_f32_16x16x4_f32(false, a1, false, b0, (short)0, acc10, false, false);
            acc11 = __builtin_amdgcn_wmma_f32_16x16x4_f32(false, a1, false, b1, (short)0, acc11, false, false);
        }
    }

    // ---- Epilogue: bias add + store (coalesced b32 across lanes) ----
    const float bv0 = bias[nbase +  0];
    const float bv1 = bias[nbase + 16];
#pragma unroll
    for (int g = 0; g < 8; ++g) {
        const int r = g + 8 * half;
        float* y0 = Y + (size_t)(m0 + r) * N_DIM + nbase;
        float* y1 = Y + (size_t)(m0 + 16 + r) * N_DIM + nbase;
        y0[ 0] = acc00[g] + bv0;
        y0[16] = acc01[g] + bv1;
        y1[ 0] = acc10[g] + bv0;
        y1[16] = acc11[g] + bv1;
    }
}

// ---------------------------------------------------------------------------
// Fused LayerNorm(512) + ReLU. One block (256 thr, 8 waves) per row.
// ---------------------------------------------------------------------------
__device__ __forceinline__ void row_moments(float2 v, float& mean, float& rstd,
                                            float* ls, float* ls2)
{
    float s  = v.x + v.y;
    float s2 = v.x * v.x + v.y * v.y;
#pragma unroll
    for (int off = 16; off > 0; off >>= 1) {
        s  += __shfl_xor(s,  off, 32);
        s2 += __shfl_xor(s2, off, 32);
    }
    const int tid  = threadIdx.x;
    const int wave = tid >> 5;
    if ((tid & 31) == 0) { ls[wave] = s; ls2[wave] = s2; }
    __syncthreads();
    if (tid == 0) {
        float ts = 0.f, ts2 = 0.f;
#pragma unroll
        for (int i = 0; i < 8; ++i) { ts += ls[i]; ts2 += ls2[i]; }
        ls[0] = ts; ls2[0] = ts2;
    }
    __syncthreads();
    mean = ls[0] * (1.0f / 512.0f);
    const float var = ls2[0] * (1.0f / 512.0f) - mean * mean;
    rstd = rsqrtf(var + 1e-5f);
}

__global__ __launch_bounds__(256)
void ln_relu_kernel(const float* __restrict__ X, const float* __restrict__ g,
                    const float* __restrict__ b, float* __restrict__ Y)
{
    __shared__ float ls[8], ls2[8];
    const int row = blockIdx.x;
    const int tid = threadIdx.x;
    const float2 v = ((const float2*)(X + (size_t)row * 512))[tid];
    float mean, rstd;
    row_moments(v, mean, rstd, ls, ls2);

    const float2 gg = ((const float2*)g)[tid];
    const float2 bb = ((const float2*)b)[tid];
    float y0 = (v.x - mean) * rstd * gg.x + bb.x;
    float y1 = (v.y - mean) * rstd * gg.y + bb.y;
    y0 = y0 > 0.f ? y0 : 0.f;
    y1 = y1 > 0.f ? y1 : 0.f;
    float2 o; o.x = y0; o.y = y1;
    ((float2*)(Y + (size_t)row * 512))[tid] = o;
}

// ---------------------------------------------------------------------------
// Fused LayerNorm + ReLU + linear(512->1) + ReLU + floor(p+0.5) -> dur[int32].
// ---------------------------------------------------------------------------
__global__ __launch_bounds__(256)
void ln_relu_linear_kernel(const float* __restrict__ X, const float* __restrict__ g,
                           const float* __restrict__ b, const float* __restrict__ lw,
                           const float* __restrict__ lb, int* __restrict__ dur)
{
    __shared__ float ls[8], ls2[8], ld[8];
    const int row = blockIdx.x;
    const int tid = threadIdx.x;
    const float2 v = ((const float2*)(X + (size_t)row * 512))[tid];
    float mean, rstd;
    row_moments(v, mean, rstd, ls, ls2);

    const float2 gg = ((const float2*)g)[tid];
    const float2 bb = ((const float2*)b)[tid];
    float y0 = (v.x - mean) * rstd * gg.x + bb.x;
    float y1 = (v.y - mean) * rstd * gg.y + bb.y;
    y0 = y0 > 0.f ? y0 : 0.f;
    y1 = y1 > 0.f ? y1 : 0.f;

    const float2 ww = ((const float2*)lw)[tid];
    float p = y0 * ww.x + y1 * ww.y;
#pragma unroll
    for (int off = 16; off > 0; off >>= 1) p += __shfl_xor(p, off, 32);
    const int wave = tid >> 5;
    if ((tid & 31) == 0) ld[wave] = p;
    __syncthreads();
    if (tid == 0) {
        float tp = 0.f;
#pragma unroll
        for (int i = 0; i < 8; ++i) tp += ld[i];
        tp += lb[0];
        tp = tp > 0.f ? tp : 0.f;                 // relu
        dur[row] = (int)floorf(tp + 0.5f);        // TINY_SLOWDOWN = 0.5
    }
}

// ---------------------------------------------------------------------------
// Inclusive cumsum of dur[4096] -> cum[4096]. Single block of 1024 threads,
// 4 elements/thread, Hillis-Steele scan over thread totals in LDS.
// ---------------------------------------------------------------------------
__global__ __launch_bounds__(1024)
void cumsum_kernel(const int* __restrict__ dur, int* __restrict__ cum)
{
    __shared__ int sh[1024];
    const int tid = threadIdx.x;
    const int4 d = ((const int4*)dur)[tid];
    const int p0 = d.x;
    const int p1 = p0 + d.y;
    const int p2 = p1 + d.z;
    const int p3 = p2 + d.w;
    sh[tid] = p3;
    __syncthreads();
    int val = p3;
    for (int off = 1; off < 1024; off <<= 1) {
        const int other = (tid >= off) ? sh[tid - off] : 0;
        __syncthreads();
        val += other;
        sh[tid] = val;
        __syncthreads();
    }
    const int excl = val - p3;
    int4 o;
    o.x = excl + p0; o.y = excl + p1; o.z = excl + p2; o.w = excl + p3;
    ((int4*)cum)[tid] = o;
}

// ---------------------------------------------------------------------------
// Length-regulator expand: frame t maps to phoneme idx = searchsorted(cum, t,
// right). Gather 512 floats (or zeros past total length). Also emits
// output_pos[t] = t+1 as int32 bits into the tail of d_out.
// ---------------------------------------------------------------------------
__global__ __launch_bounds__(256)
void expand_kernel(const float* __restrict__ enc, const int* __restrict__ cum,
                   float* __restrict__ out, int* __restrict__ pos)
{
    const int t   = blockIdx.x;
    const int tid = threadIdx.x;
    const int total = cum[S_LEN - 1];

    // searchsorted(cum, t, side='right'): first i with cum[i] > t
    int lo = 0, hi = S_LEN;
    while (lo < hi) {
        const int mid = (lo + hi) >> 1;
        if (cum[mid] <= t) lo = mid + 1; else hi = mid;
    }
    int idx = lo < (S_LEN - 1) ? lo : (S_LEN - 1);
    const bool valid = t < total;

    float2 v = ((const float2*)(enc + (size_t)idx * D_DIM))[tid];
    if (!valid) { v.x = 0.f; v.y = 0.f; }
    ((float2*)(out + (size_t)t * D_DIM))[tid] = v;

    if (tid == 0) pos[t] = t + 1;
}

// ---------------------------------------------------------------------------
extern "C" void kernel_launch(void* const* d_in, const int* in_sizes, int n_in,
                              void* d_out, int out_size, void* d_ws, size_t ws_size,
                              hipStream_t stream)
{
    (void)in_sizes; (void)n_in; (void)out_size; (void)ws_size;

    const float* enc = (const float*)d_in[0];   // [1,4096,512]
    const float* c1w = (const float*)d_in[1];   // [512,512,3]
    const float* c1b = (const float*)d_in[2];   // [512]
    const float* g1  = (const float*)d_in[3];
    const float* b1  = (const float*)d_in[4];
    const float* c2w = (const float*)d_in[5];   // [512,512,3]
    const float* c2b = (const float*)d_in[6];
    const float* g2  = (const float*)d_in[7];
    const float* b2  = (const float*)d_in[8];
    const float* lw  = (const float*)d_in[9];   // [1,512]
    const float* lb  = (const float*)d_in[10];  // [1]

    const size_t act = (size_t)S_LEN * N_DIM;            // 2M floats
    const size_t wsz = (size_t)KW * D_DIM * N_DIM;       // 786432 floats
    float* buf0 = (float*)d_ws;                          // [4096,512]
    float* buf1 = buf0 + act;                            // [4096,512]
    float* wt1  = buf1 + act;                            // [3,512,512]
    float* wt2  = wt1 + wsz;                             // [3,512,512]
    int*   dur  = (int*)(wt2 + wsz);                     // [4096]
    int*   cum  = dur + S_LEN;                           // [4096]

    const int rp_blocks = (int)(wsz / 256);
    repack_w_kernel<<<rp_blocks, 256, 0, stream>>>(c1w, wt1);
    repack_w_kernel<<<rp_blocks, 256, 0, stream>>>(c2w, wt2);

    const int gemm_blocks = (S_LEN / M_BLK) * 2;         // 256 blocks
    conv1d_gemm_wmma<<<gemm_blocks, 256, 0, stream>>>(enc, wt1, c1b, buf0);
    ln_relu_kernel<<<S_LEN, 256, 0, stream>>>(buf0, g1, b1, buf1);
    conv1d_gemm_wmma<<<gemm_blocks, 256, 0, stream>>>(buf1, wt2, c2b, buf0);
    ln_relu_linear_kernel<<<S_LEN, 256, 0, stream>>>(buf0, g2, b2, lw, lb, dur);
    cumsum_kernel<<<1, 1024, 0, stream>>>(dur, cum);

    float* out = (float*)d_out;                               // [16384,512] f32
    int*   pos = (int*)(out + (size_t)MAX_OUT * D_DIM);       // [16384] int32 bits
    expand_kernel<<<MAX_OUT, 256, 0, stream>>>(enc, cum, out, pos);
}